// TensorProductContraction_50199577756234
// MI455X (gfx1250) — compile-verified
//
#include <hip/hip_runtime.h>

typedef float v2f __attribute__((ext_vector_type(2)));
typedef float v8f __attribute__((ext_vector_type(8)));

#define D_IN    1536
#define D_OUT   16384
#define BROWS   16
#define XSTRIDE 1540   // 1536 + 4 pad -> conflict-free LDS A reads (64 banks)

__device__ __forceinline__ v8f wmma4(v2f a, v2f b, v8f c) {
  // D = A(16x4 f32) * B(4x16 f32) + C(16x16 f32)
  return __builtin_amdgcn_wmma_f32_16x16x4_f32(
      /*neg_a=*/false, a, /*neg_b=*/false, b,
      /*c_mod=*/(short)0, c, /*reuse_a=*/false, /*reuse_b=*/false);
}

// K=128 GEMM for one irrep section with D components.
// w: [128, ldw] row-major; column tile starts at n0.
// xrow: &xs[(lane&15)*XSTRIDE]; xoff: section column offset in x.
template<int D>
__device__ __forceinline__ void gemm_sec(const float* __restrict__ w, int ldw,
                                         int n0, const float* xrow, int xoff,
                                         int nl, int kh, v8f* acc) {
  const float* wp = w + (size_t)kh * ldw + n0 + nl;   // B[k][n]: n=lane&15, k=2*(lane>>4)+vgpr
  #pragma unroll 4
  for (int s = 0; s < 32; ++s) {
    const int k = 4 * s + kh;
    v2f b;
    b.x = wp[(size_t)(4 * s) * ldw];
    b.y = wp[(size_t)(4 * s + 1) * ldw];
    #pragma unroll
    for (int c = 0; c < D; ++c) {
      v2f a;                                          // A[m][k]: m=lane&15
      a.x = xrow[xoff + k * D + c];
      a.y = xrow[xoff + (k + 1) * D + c];
      acc[c] = wmma4(a, b, acc[c]);
    }
  }
}

__global__ __launch_bounds__(256)
void tp_cg_fused_kernel(const float* __restrict__ x,
                        const float* __restrict__ w0e,
                        const float* __restrict__ w1o,
                        const float* __restrict__ w1e,
                        const float* __restrict__ w2e,
                        float* __restrict__ out) {
  __shared__ float xs[BROWS * XSTRIDE];
  const int row0 = blockIdx.x * BROWS;

  // Stage x tile (16 x 1536) into LDS, row-major with padded stride.
  for (int i = threadIdx.x; i < BROWS * D_IN; i += 256) {
    const int r = i / D_IN;
    const int c = i - r * D_IN;
    xs[r * XSTRIDE + c] = x[(size_t)(row0 + r) * D_IN + c];
  }
  __syncthreads();

  const int lane = threadIdx.x & 31;
  const int wave = threadIdx.x >> 5;
  const int nl = lane & 15;        // A-row / B-col / C-col index
  const int h2 = lane >> 4;        // lane half
  const int kh = h2 << 1;          // k sub-offset within a 4-wide K step
  const float* xrow = &xs[nl * XSTRIDE];

  const float i3  = 0.57735026918962576f;  // 1/sqrt(3)
  const float i6  = 0.40824829046386302f;  // 1/sqrt(6)
  const float i10 = 0.31622776601683793f;  // 1/sqrt(10)
  const float i30 = 0.18257418583505537f;  // 1/sqrt(30)

  const v8f zero = {0.f, 0.f, 0.f, 0.f, 0.f, 0.f, 0.f, 0.f};

  // 256 jobs: [0,64) block3 fused (0e-lo,1e,2e), [64,128) block2 (1o-lo),
  //           [128,192) block1 (1o-hi), [192,256) block0 (0e-hi).
  for (int job = wave; job < 256; job += 8) {
    const int typ  = job >> 6;
    const int tile = job & 63;

    if (typ == 0) {
      // ---- block3: out[7168 + m*288 + lc*96 + nn*3 + rc] ----
      const int n0 = tile * 16;            // v in [0,1024)
      v8f aS[1] = {zero};
      gemm_sec<1>(w0e, 2048, n0, xrow, 0,   nl, kh, aS);
      v8f aE[3] = {zero, zero, zero};
      gemm_sec<3>(w1e, 1024, n0, xrow, 512, nl, kh, aE);
      v8f aG[5] = {zero, zero, zero, zero, zero};
      gemm_sec<5>(w2e, 1024, n0, xrow, 896, nl, kh, aG);

      const int v = n0 + nl;
      const int m = v >> 5, nn = v & 31;
      const size_t cbase = 7168 + (size_t)m * 288 + (size_t)nn * 3;
      #pragma unroll
      for (int j = 0; j < 8; ++j) {
        const float s  = aS[0][j] * i3;
        const float e0 = aE[0][j] * i6, e1 = aE[1][j] * i6, e2 = aE[2][j] * i6;
        const float g0 = aG[0][j] * i10, g1 = aG[1][j] * i10;
        const float g2 = aG[2][j] * i30;
        const float g3 = aG[3][j] * i10, g4 = aG[4][j] * i10;
        float* p = out + (size_t)(row0 + j + 8 * h2) * D_OUT + cbase;
        p[0]   = s - g2 - g4;        // O00
        p[1]   =  e2 + g1;           // O01
        p[2]   = -e1 + g0;           // O02
        p[96]  = -e2 + g1;           // O10
        p[97]  = s + 2.f * g2;       // O11
        p[98]  =  e0 + g3;           // O12
        p[192] =  e1 + g0;           // O20
        p[193] = -e0 + g3;           // O21
        p[194] = s - g2 + g4;        // O22
      }
    } else if (typ == 1) {
      // ---- block2: out[4096 + m*96 + lc*32 + nn] = y1o[v<1024, lc]/sqrt3 ----
      const int n0 = tile * 16;
      v8f a1[3] = {zero, zero, zero};
      gemm_sec<3>(w1o, 2048, n0, xrow, 128, nl, kh, a1);
      const int v = n0 + nl;
      const int m = v >> 5, nn = v & 31;
      const size_t cbase = 4096 + (size_t)m * 96 + nn;
      #pragma unroll
      for (int j = 0; j < 8; ++j) {
        float* p = out + (size_t)(row0 + j + 8 * h2) * D_OUT + cbase;
        p[0]  = a1[0][j] * i3;
        p[32] = a1[1][j] * i3;
        p[64] = a1[2][j] * i3;
      }
    } else if (typ == 2) {
      // ---- block1: out[1024 + m*96 + nn*3 + rc] = y1o[v>=1024, rc]/sqrt3 ----
      const int n0 = 1024 + tile * 16;
      v8f a1[3] = {zero, zero, zero};
      gemm_sec<3>(w1o, 2048, n0, xrow, 128, nl, kh, a1);
      const int t = n0 + nl - 1024;
      const int m = t >> 5, nn = t & 31;
      const size_t cbase = 1024 + (size_t)m * 96 + (size_t)nn * 3;
      #pragma unroll
      for (int j = 0; j < 8; ++j) {
        float* p = out + (size_t)(row0 + j + 8 * h2) * D_OUT + cbase;
        p[0] = a1[0][j] * i3;
        p[1] = a1[1][j] * i3;
        p[2] = a1[2][j] * i3;
      }
    } else {
      // ---- block0: out[v-1024] = y0e[v>=1024] ----
      const int n0 = 1024 + tile * 16;
      v8f a0[1] = {zero};
      gemm_sec<1>(w0e, 2048, n0, xrow, 0, nl, kh, a0);
      const size_t cbase = (size_t)(n0 - 1024) + nl;
      #pragma unroll
      for (int j = 0; j < 8; ++j) {
        out[(size_t)(row0 + j + 8 * h2) * D_OUT + cbase] = a0[0][j];
      }
    }
  }
}

extern "C" void kernel_launch(void* const* d_in, const int* in_sizes, int n_in,
                              void* d_out, int out_size, void* d_ws, size_t ws_size,
                              hipStream_t stream) {
  const float* x   = (const float*)d_in[0];
  const float* w0e = (const float*)d_in[1];
  const float* w1o = (const float*)d_in[2];
  const float* w1e = (const float*)d_in[3];
  const float* w2e = (const float*)d_in[4];
  float* out = (float*)d_out;

  dim3 grid(8192 / BROWS);   // 512 row tiles
  dim3 block(256);           // 8 wave32 waves
  tp_cg_fused_kernel<<<grid, block, 0, stream>>>(x, w0e, w1o, w1e, w2e, out);
}